// MultiPositiveLoss_8761733284104
// MI455X (gfx1250) — compile-verified
//
#include <hip/hip_runtime.h>

#define C_CLASSES 2048
#define BATCH_N   32768

typedef __attribute__((ext_vector_type(2))) float v2f;
typedef __attribute__((ext_vector_type(8))) float v8f;

__global__ void MPL_init_ws(double* __restrict__ acc, int* __restrict__ count) {
    if (threadIdx.x == 0) { *acc = 0.0; *count = 0; }
}

// One row per thread. For t != 0 the negative set is {0}, so the loss needs only
// x[row,0] and x[row,t]. Rows with t == 0 go to a worklist for the slow kernel.
// 32 per-lane partial losses are reduced with one V_WMMA_F32_16X16X4_F32 vs an
// all-ones B: D[m,n] = partial[m] + partial[m+16]; summing the 8 D VGPRs gives
// lane0 = sum(M=0..7, N=0), lane16 = sum(M=8..15, N=0).
__global__ __launch_bounds__(256) void MPL_fast_rows(const float* __restrict__ x,
                                                     const int* __restrict__ tgt,
                                                     double* __restrict__ acc,
                                                     int* __restrict__ count,
                                                     int* __restrict__ list) {
    const int row = blockIdx.x * 256 + threadIdx.x;   // grid exactly covers BATCH_N
    const int t = tgt[row];
    float partial = 0.0f;
    if (t != 0) {
        const float* px = x + (size_t)row * C_CLASSES;
        const float x0 = px[0];
        const float xt = px[t];
        partial = __logf(__expf(x0) + __expf(xt)) - xt;
    } else {
        const int pos = atomicAdd(count, 1);
        list[pos] = row;
    }

    // Wave-level reduction on the matrix pipe (EXEC is all ones here).
    v2f a; a[0] = partial; a[1] = 0.0f;     // A 16x4: lanes0-15 K=0, lanes16-31 K=2; K=1,3 zero
    v2f b; b[0] = 1.0f;    b[1] = 1.0f;     // B 4x16 all ones
    v8f c = {};
    v8f d = __builtin_amdgcn_wmma_f32_16x16x4_f32(
        /*neg_a=*/false, a, /*neg_b=*/false, b,
        /*c_mod=*/(short)0, c, /*reuse_a=*/false, /*reuse_b=*/false);
    float s = d[0] + d[1] + d[2] + d[3] + d[4] + d[5] + d[6] + d[7];
    const float hi = __shfl(s, 16, 32);     // lane16 holds sum(M=8..15)
    if ((threadIdx.x & 31) == 0) {
        atomicAdd(acc, (double)(s + hi));   // one f64 atomic per wave
    }
}

// Full log-sum-exp for the ~B/C rows whose target is 0.
__global__ __launch_bounds__(256) void MPL_zero_target_rows(const float* __restrict__ x,
                                                            const int* __restrict__ count,
                                                            const int* __restrict__ list,
                                                            double* __restrict__ acc) {
    __shared__ float wsum[8];
    const int n = *count;
    for (int i = blockIdx.x; i < n; i += gridDim.x) {
        const int row = list[i];
        const float4* px = (const float4*)(x + (size_t)row * C_CLASSES);
        float s = 0.0f;
        #pragma unroll
        for (int j = 0; j < (C_CLASSES / 4) / 256; ++j) {
            const float4 v = px[threadIdx.x + j * 256];
            s += __expf(v.x) + __expf(v.y) + __expf(v.z) + __expf(v.w);
        }
        // wave32 reduce
        #pragma unroll
        for (int off = 16; off > 0; off >>= 1) s += __shfl_down(s, off, 32);
        if ((threadIdx.x & 31) == 0) wsum[threadIdx.x >> 5] = s;
        __syncthreads();
        if (threadIdx.x == 0) {
            float tot = 0.0f;
            #pragma unroll
            for (int w = 0; w < 8; ++w) tot += wsum[w];
            const float x0 = x[(size_t)row * C_CLASSES];
            atomicAdd(acc, (double)(__logf(tot) - x0));
        }
        __syncthreads();   // protect wsum before next worklist entry
    }
}

__global__ void MPL_finalize(const double* __restrict__ acc, float* __restrict__ out) {
    if (threadIdx.x == 0) out[0] = (float)(*acc * (1.0 / (double)BATCH_N));
}

extern "C" void kernel_launch(void* const* d_in, const int* in_sizes, int n_in,
                              void* d_out, int out_size, void* d_ws, size_t ws_size,
                              hipStream_t stream) {
    const float* x   = (const float*)d_in[0];   // [BATCH_N, C_CLASSES] fp32
    const int*   tgt = (const int*)d_in[1];     // [BATCH_N] int32
    float*       out = (float*)d_out;           // scalar fp32

    double* acc   = (double*)d_ws;                    // 8 B
    int*    count = (int*)((char*)d_ws + 8);          // 4 B (+4 pad)
    int*    list  = (int*)((char*)d_ws + 16);         // up to 4*BATCH_N bytes

    MPL_init_ws<<<1, 64, 0, stream>>>(acc, count);
    MPL_fast_rows<<<BATCH_N / 256, 256, 0, stream>>>(x, tgt, acc, count, list);
    MPL_zero_target_rows<<<64, 256, 0, stream>>>(x, count, list, acc);
    MPL_finalize<<<1, 64, 0, stream>>>(acc, out);
}